// LinearBlockSparseAttention_20237885899056
// MI455X (gfx1250) — compile-verified
//
#include <hip/hip_runtime.h>
#include <cmath>

// ---------------------------------------------------------------------------
// LinearBlockSparseAttention for MI455X (gfx1250)
//   - fp32 WMMA (V_WMMA_F32_16X16X4_F32) for all matrix math (matches fp32 ref)
//   - GEMM stages use the Tensor Data Mover (TENSOR_LOAD_TO_LDS) with double-
//     buffered LDS staging, padded via the D# pad fields for conflict-free
//     fragment reads.
// ---------------------------------------------------------------------------

#define DIM      1024
#define HEADS    16
#define HEAD_DIM 64
#define BLOCK    32
#define FEAT     128
#define EPS      1e-8f

typedef float v2f __attribute__((ext_vector_type(2)));
typedef float v8f __attribute__((ext_vector_type(8)));
typedef unsigned int u32x4 __attribute__((ext_vector_type(4)));
typedef int i32x4 __attribute__((ext_vector_type(4)));
typedef int i32x8 __attribute__((ext_vector_type(8)));

__device__ __forceinline__ v8f wmma4(v2f a, v2f b, v8f c) {
    // D = A(16x4 f32) * B(4x16 f32) + C(16x16 f32)
    return __builtin_amdgcn_wmma_f32_16x16x4_f32(
        /*neg_a=*/false, a, /*neg_b=*/false, b,
        /*c_mod=*/(short)0, c, /*reuse_a=*/false, /*reuse_b=*/false);
}

// elu(x)+1 : x>0 ? x+1 : exp(x)
__device__ __forceinline__ float elup1(float x) {
    return x > 0.0f ? x + 1.0f : __expf(x);
}

// ---------------------------------------------------------------------------
// TDM: issue a 2D tile load Global -> LDS.
//   tile_d0 elements contiguous in memory (x), tile_d1 lines (y),
//   line stride = stride_elems (fp32 elements). Optional LDS padding via the
//   D# group1 pad fields (ISA 8.4: interval 2^(code+1) dwords, amount
//   (code+1) dwords). Tensor dims are set equal to tile dims (tiles never go
//   out of bounds in this problem).
// ---------------------------------------------------------------------------
__device__ __forceinline__ void tdm_load_2d_f32(
    unsigned lds_byte_addr, const void* gptr,
    unsigned tile_d0, unsigned tile_d1,
    unsigned long long stride_elems,
    unsigned pad_interval_code, unsigned pad_amount_code, int do_pad)
{
    unsigned long long ga = (unsigned long long)(uintptr_t)gptr;

    u32x4 g0;
    g0.x = 1u;                                           // count=1, user D#
    g0.y = lds_byte_addr;                                // lds_addr (bytes)
    g0.z = (unsigned)(ga & 0xFFFFFFFFull);               // global_addr[31:0]
    g0.w = (unsigned)((ga >> 32) & 0x01FFFFFFull)        // global_addr[56:32]
         | (2u << 30);                                   // type = 2 ("image")

    i32x8 g1;
    unsigned dw0 = (2u << 16);                           // data_size = 4 bytes
    if (do_pad)
        dw0 |= (1u << 20)                                // pad_enable
             | (pad_interval_code << 22)                 // pad_interval
             | (pad_amount_code << 25);                  // pad_amount
    g1[0] = (int)dw0;                                    // wg_mask=0 (no cluster)
    g1[1] = (int)((tile_d0 & 0xFFFFu) << 16);            // tensor_dim0[15:0]
    g1[2] = (int)(((tile_d0 >> 16) & 0xFFFFu)            // tensor_dim0[31:16]
                | ((tile_d1 & 0xFFFFu) << 16));          // tensor_dim1[15:0]
    g1[3] = (int)(((tile_d1 >> 16) & 0xFFFFu)            // tensor_dim1[31:16]
                | ((tile_d0 & 0xFFFFu) << 16));          // tile_dim0
    g1[4] = (int)(tile_d1 & 0xFFFFu);                    // tile_dim1; tile_dim2=0
    g1[5] = (int)(unsigned)(stride_elems & 0xFFFFFFFFull);   // dim0_stride[31:0]
    g1[6] = (int)(unsigned)((stride_elems >> 32) & 0xFFFFull);// dim0_stride[47:32]
    g1[7] = 0;                                           // dim1_stride unused (2D)

    i32x4 gz4 = {0, 0, 0, 0};                            // groups 2/3 unused (2D)
    i32x8 gz8 = {0, 0, 0, 0, 0, 0, 0, 0};                // extra group unused
    __builtin_amdgcn_tensor_load_to_lds(g0, g1, gz4, gz4, gz8, /*cpol=*/0);
}

// ---------------------------------------------------------------------------
// Generic row-major GEMM: C[M,N] = A[M,K] * B[K,N] (+ bias).
// 256 threads = 8 waves; each wave owns a 32x64 region as 2x4 WMMA tiles.
// Workgroup tile: 128(M) x 128(N); K consumed in 32-wide chunks staged into
// LDS by the TDM with double buffering.
//
// LDS (dynamic): A buf = 128 rows * 34 floats (32 + 2 pad), B buf = 32 rows *
// 144 floats (128 + 16 pad); two buffers each -> 71,680 bytes.
//   A frag (b64): addr stride 34 across lanes -> conflict-free.
//   B frag (b32): half-wave row offset 2*144 = 288 = 32 (mod 64) banks ->
//                 the two 16-lane groups hit disjoint bank windows.
// ---------------------------------------------------------------------------
#define KC        32
#define AS_STRIDE 34            // floats per staged A row
#define BS_STRIDE 144           // floats per staged B row
#define A_BUF_FL  (128 * AS_STRIDE)   // 4352 floats
#define B_BUF_FL  (KC * BS_STRIDE)    // 4608 floats
#define GEMM_LDS_FL (2 * (A_BUF_FL + B_BUF_FL))   // 17920 floats = 71680 B

template <bool ADD_BIAS>
__global__ __launch_bounds__(256) void gemm_wmma_f32(
    const float* __restrict__ A, const float* __restrict__ Bm,
    const float* __restrict__ bias, float* __restrict__ C,
    int M, int N, int K)
{
    extern __shared__ float smem[];
    const unsigned lds_base = (unsigned)(uintptr_t)smem;   // LDS byte offset

    const int tid = threadIdx.x;
    const int ln  = tid & 31;          // lane in wave32
    const int wv  = tid >> 5;          // wave 0..7
    const int hl  = ln >> 4;           // 0 = lanes 0-15, 1 = lanes 16-31
    const int l   = ln & 15;

    const int wm = wv >> 1;            // 0..3  (M dir, 32 rows each)
    const int wn = wv & 1;             // 0..1  (N dir, 64 cols each)

    const int rowBase0 = blockIdx.x * 128;          // WG tile origin
    const int colBase0 = blockIdx.y * 128;
    const int rowBase  = rowBase0 + wm * 32;
    const int colBase  = colBase0 + wn * 64;

    // LDS buffer offsets (floats)
    const unsigned aOff[2] = {0u, (unsigned)(A_BUF_FL + B_BUF_FL)};
    const unsigned bOff[2] = {(unsigned)A_BUF_FL,
                              (unsigned)(2 * A_BUF_FL + B_BUF_FL)};

    // --- stage chunk 0 (buffer 0) via TDM, wave 0 only ---
    if (wv == 0) {
        // A chunk: 128 rows x 32 k, row stride K; pad 2 dwords / 32 dwords
        tdm_load_2d_f32(lds_base + aOff[0] * 4,
                        A + (size_t)rowBase0 * K,
                        /*d0=*/KC, /*d1=*/128, (unsigned long long)K,
                        /*interval 32dw=*/4, /*amount 2dw=*/1, 1);
        // B chunk: 32 k-rows x 128 cols, row stride N; pad 16 dwords / 128
        tdm_load_2d_f32(lds_base + bOff[0] * 4,
                        Bm + colBase0,
                        /*d0=*/128, /*d1=*/KC, (unsigned long long)N,
                        /*interval 128dw=*/6, /*amount 16dw=*/15, 1);
        __builtin_amdgcn_s_wait_tensorcnt(0);
    }
    __syncthreads();

    v8f acc[2][4] = {};
    int p = 0;

    for (int kc = 0; kc < K; kc += KC) {
        // Prefetch next chunk into the other buffer while computing this one.
        const bool more = (kc + KC) < K;
        if (more && wv == 0) {
            tdm_load_2d_f32(lds_base + aOff[p ^ 1] * 4,
                            A + (size_t)rowBase0 * K + (kc + KC),
                            KC, 128, (unsigned long long)K, 4, 1, 1);
            tdm_load_2d_f32(lds_base + bOff[p ^ 1] * 4,
                            Bm + (size_t)(kc + KC) * N + colBase0,
                            128, KC, (unsigned long long)N, 6, 15, 1);
        }

        const float* As  = smem + aOff[p];
        const float* Bs  = smem + bOff[p];
        const float* a0p = As + (size_t)(wm * 32 + l)      * AS_STRIDE + 2 * hl;
        const float* a1p = As + (size_t)(wm * 32 + 16 + l) * AS_STRIDE + 2 * hl;
        const float* bp  = Bs + (size_t)(2 * hl) * BS_STRIDE + wn * 64 + l;

        #pragma unroll
        for (int kk = 0; kk < KC; kk += 4) {
            v2f a0 = *(const v2f*)(a0p + kk);   // 8B aligned: 34*row + even
            v2f a1 = *(const v2f*)(a1p + kk);
            const float* br0 = bp + (size_t)kk * BS_STRIDE;
            const float* br1 = br0 + BS_STRIDE;
            #pragma unroll
            for (int ni = 0; ni < 4; ++ni) {
                v2f b;
                b.x = br0[ni * 16];
                b.y = br1[ni * 16];
                acc[0][ni] = wmma4(a0, b, acc[0][ni]);
                acc[1][ni] = wmma4(a1, b, acc[1][ni]);
            }
        }

        if (more && wv == 0) __builtin_amdgcn_s_wait_tensorcnt(0);
        __syncthreads();
        p ^= 1;
    }

    // C layout: VGPR i -> row (i + 8*hl), col = lane&15 within the 16x16 tile
    #pragma unroll
    for (int mi = 0; mi < 2; ++mi) {
        const int r0 = rowBase + mi * 16 + hl * 8;
        #pragma unroll
        for (int ni = 0; ni < 4; ++ni) {
            const int c = colBase + ni * 16 + l;
            const float bv = ADD_BIAS ? bias[c] : 0.0f;
            #pragma unroll
            for (int i = 0; i < 8; ++i) {
                C[(size_t)(r0 + i) * N + c] = acc[mi][ni][i] + bv;
            }
        }
    }
}

// ---------------------------------------------------------------------------
// Per-(block, head) linear attention. One workgroup = 256 threads = 8 waves
// per (token-block g, head h). Token rows r0 = g*32 .. +31.
//
// LDS (dynamic, padded against bank conflicts):
//   s_qf [32][129], s_kf [32][129], s_v [32][65], s_kv [128][65],
//   s_ksum [128], s_z [32]   -> 18816 floats = 75264 bytes
// ---------------------------------------------------------------------------
__global__ __launch_bounds__(256) void attn_block_kernel(
    const float* __restrict__ qkv,   // [MT, 3*DIM]
    const float* __restrict__ proj,  // [HEADS, HEAD_DIM, FEAT]
    float* __restrict__ attn)        // [MT, DIM]
{
    extern __shared__ float smem[];
    float* s_qf   = smem;                       // 32*129
    float* s_kf   = s_qf  + 32 * 129;           // 32*129
    float* s_v    = s_kf  + 32 * 129;           // 32*65
    float* s_kv   = s_v   + 32 * 65;            // 128*65
    float* s_ksum = s_kv  + 128 * 65;           // 128
    float* s_z    = s_ksum + 128;               // 32

    const int tid = threadIdx.x;
    const int ln  = tid & 31;
    const int wv  = tid >> 5;
    const int hl  = ln >> 4;
    const int l   = ln & 15;

    const int g  = blockIdx.x;           // token block
    const int h  = blockIdx.y;           // head
    const int r0 = g * BLOCK;            // first token row

    const float* projh = proj + (size_t)h * HEAD_DIM * FEAT;

    // ---- Phase A: stage v into LDS; compute qf = elup1(q@proj), kf likewise
    for (int idx = tid; idx < BLOCK * HEAD_DIM; idx += 256) {
        const int s = idx >> 6, e = idx & 63;
        s_v[s * 65 + e] =
            qkv[(size_t)(r0 + s) * (3 * DIM) + 2 * DIM + h * HEAD_DIM + e];
    }

    // 32 tiles: {q,k} x 2 M-tiles x 8 N-tiles; 4 per wave
    #pragma unroll
    for (int t = 0; t < 4; ++t) {
        const int T   = t * 8 + wv;          // 0..31
        const int mat = T >> 4;              // 0=q, 1=k
        const int tm  = (T >> 3) & 1;        // 0..1
        const int tn  = T & 7;               // 0..7
        const float* Abase =
            qkv + (size_t)r0 * (3 * DIM) + mat * DIM + h * HEAD_DIM;
        const float* arow = Abase + (size_t)(tm * 16 + l) * (3 * DIM) + 2 * hl;

        v8f acc = {};
        for (int k0 = 0; k0 < HEAD_DIM; k0 += 4) {
            v2f a = *(const v2f*)(arow + k0);
            const int ka = k0 + 2 * hl;
            v2f b;
            b.x = projh[(size_t)ka       * FEAT + tn * 16 + l];
            b.y = projh[(size_t)(ka + 1) * FEAT + tn * 16 + l];
            acc = wmma4(a, b, acc);
        }
        float* dst = mat ? s_kf : s_qf;
        #pragma unroll
        for (int i = 0; i < 8; ++i) {
            dst[(tm * 16 + hl * 8 + i) * 129 + tn * 16 + l] = elup1(acc[i]);
        }
    }
    __syncthreads();

    // ---- Phase B: kv[f][d] = sum_s kf[s][f] * v[s][d]   (M=128, N=64, K=32)
    #pragma unroll
    for (int t = 0; t < 4; ++t) {
        const int T  = t * 8 + wv;           // 0..31
        const int tm = T >> 2;               // 0..7 (feature tiles)
        const int tn = T & 3;                // 0..3 (head-dim tiles)
        v8f acc = {};
        for (int k0 = 0; k0 < BLOCK; k0 += 4) {
            const int ka = k0 + 2 * hl;
            v2f a, b;
            a.x = s_kf[ka       * 129 + tm * 16 + l];
            a.y = s_kf[(ka + 1) * 129 + tm * 16 + l];
            b.x = s_v[ka       * 65 + tn * 16 + l];
            b.y = s_v[(ka + 1) * 65 + tn * 16 + l];
            acc = wmma4(a, b, acc);
        }
        #pragma unroll
        for (int i = 0; i < 8; ++i) {
            s_kv[(tm * 16 + hl * 8 + i) * 65 + tn * 16 + l] = acc[i];
        }
    }
    __syncthreads();

    // ---- Phase C: ksum[f] = sum_s kf[s][f]; z[s] = 1/(qf[s]·ksum + eps)
    if (tid < FEAT) {
        float s = 0.0f;
        #pragma unroll
        for (int r = 0; r < BLOCK; ++r) s += s_kf[r * 129 + tid];
        s_ksum[tid] = s;
    }
    __syncthreads();
    if (tid < BLOCK) {
        float acc = 0.0f;
        for (int f = 0; f < FEAT; ++f) acc += s_qf[tid * 129 + f] * s_ksum[f];
        s_z[tid] = 1.0f / (acc + EPS);
    }
    __syncthreads();

    // ---- Phase D: num = qf @ kv (M=32, N=64, K=128); scale by z; store.
    {
        const int tm = wv >> 2;              // 0..1
        const int tn = wv & 3;               // 0..3
        v8f acc = {};
        for (int k0 = 0; k0 < FEAT; k0 += 4) {
            const int ka = k0 + 2 * hl;
            v2f a, b;
            a.x = s_qf[(tm * 16 + l) * 129 + ka];
            a.y = s_qf[(tm * 16 + l) * 129 + ka + 1];
            b.x = s_kv[ka       * 65 + tn * 16 + l];
            b.y = s_kv[(ka + 1) * 65 + tn * 16 + l];
            acc = wmma4(a, b, acc);
        }
        #pragma unroll
        for (int i = 0; i < 8; ++i) {
            const int s = tm * 16 + hl * 8 + i;
            attn[(size_t)(r0 + s) * DIM + h * HEAD_DIM + tn * 16 + l] =
                acc[i] * s_z[s];
        }
    }
}

// ---------------------------------------------------------------------------
// Launch
// ---------------------------------------------------------------------------
extern "C" void kernel_launch(void* const* d_in, const int* in_sizes, int n_in,
                              void* d_out, int out_size, void* d_ws, size_t ws_size,
                              hipStream_t stream)
{
    const float* x    = (const float*)d_in[0];   // [b, n, DIM]
    const float* Wqkv = (const float*)d_in[1];   // [DIM, 3*DIM]
    const float* proj = (const float*)d_in[2];   // [HEADS, HEAD_DIM, FEAT]
    const float* Wout = (const float*)d_in[3];   // [DIM, DIM]
    const float* bout = (const float*)d_in[4];   // [DIM]
    float* out = (float*)d_out;

    const int MT = in_sizes[0] / DIM;            // total tokens = b*n = 16384

    float* qkv  = (float*)d_ws;                  // [MT, 3*DIM]  (192 MB)
    float* attn = qkv + (size_t)MT * 3 * DIM;    // [MT, DIM]    ( 64 MB)

    dim3 blk(256);
    const size_t gemm_lds = (size_t)GEMM_LDS_FL * sizeof(float);   // 71680 B

    // Stage 1: qkv = x @ Wqkv
    gemm_wmma_f32<false><<<dim3(MT / 128, (3 * DIM) / 128), blk, gemm_lds,
                           stream>>>(x, Wqkv, nullptr, qkv, MT, 3 * DIM, DIM);

    // Stage 2: per-(block, head) linear attention
    const size_t attn_lds = (size_t)(32 * 129 + 32 * 129 + 32 * 65 +
                                     128 * 65 + 128 + 32) * sizeof(float);
    attn_block_kernel<<<dim3(MT / BLOCK, HEADS), blk, attn_lds, stream>>>(
        qkv, proj, attn);

    // Stage 3: out = attn @ Wout + bout
    gemm_wmma_f32<true><<<dim3(MT / 128, DIM / 128), blk, gemm_lds, stream>>>(
        attn, Wout, bout, out, MT, DIM, DIM);
}